// BezierGlyph_20169166422516
// MI455X (gfx1250) — compile-verified
//
#include <hip/hip_runtime.h>

// --- problem constants (from reference) ---
#define SIZE        512
#define NPIX        (SIZE * SIZE)           // 262144 pixels
#define N_STROKES   16
#define N_SAMPLES   32
#define NPTS        (N_STROKES * N_SAMPLES) // 512 curve samples
#define SHARP       256.0f                  // N_SAMPLES * 8
#define SW          0.04f                   // stroke width
#define LOG2E       1.44269504088896340736f
#define LN2         0.69314718055994530942f
#define FSCALE      (SHARP * LOG2E)         // fold exp scale into the GEMM
#define EPS         0.5f                    // C-operand bias: keeps D >= 0

#define WAVES_PER_BLOCK 8
#define BLOCK           (WAVES_PER_BLOCK * 32)

typedef float v2f __attribute__((ext_vector_type(2)));
typedef float v8f __attribute__((ext_vector_type(8)));

__device__ __forceinline__ float clip01(float v) {
    return fminf(fmaxf(v, 0.0f), 1.0f);
}

__global__ __launch_bounds__(BLOCK) void BezierGlyph_wmma_kernel(
    const float* __restrict__ cp,   // control points: 16 x 4 x 2 f32
    const float* __restrict__ pg,   // pixel grid:    NPIX x 2 f32
    float* __restrict__ out,        // NPIX f32
    int npix)
{
    // ---- Stage 1: Bezier samples -> LDS, pre-scaled & pre-laid-out for A ----
    // spts[m] = { f*sx, f*sy, f*(sx^2+sy^2), f }  (f = SHARP*log2e)
    // lanes 0-15 consume .xy (K rows 0,1), lanes 16-31 consume .zw (K rows 2,3)
    __shared__ float4 spts[NPTS];   // 8 KB
    for (int t = (int)threadIdx.x; t < NPTS; t += BLOCK) {
        int s = t >> 5;          // stroke 0..15
        int j = t & 31;          // sample 0..31
        float tt = (float)j * (1.0f / (float)(N_SAMPLES - 1));
        float mt = 1.0f - tt;
        const float* c = cp + s * 8;
        float p0x = clip01(c[0]), p0y = clip01(c[1]);
        float p1x = clip01(c[2]), p1y = clip01(c[3]);
        float p2x = clip01(c[4]), p2y = clip01(c[5]);
        float p3x = clip01(c[6]), p3y = clip01(c[7]);
        float w0 = mt * mt * mt;
        float w1 = 3.0f * mt * mt * tt;
        float w2 = 3.0f * mt * tt * tt;
        float w3 = tt * tt * tt;
        float sx = w0 * p0x + w1 * p1x + w2 * p2x + w3 * p3x;
        float sy = w0 * p0y + w1 * p1y + w2 * p2y + w3 * p3y;
        float sn = sx * sx + sy * sy;
        spts[t] = make_float4(FSCALE * sx, FSCALE * sy, FSCALE * sn, FSCALE);
    }
    __syncthreads();

    // ---- Stage 2: one wave per 16-pixel column tile ----
    const int lane = (int)threadIdx.x & 31;
    const int half = lane >> 4;          // 0: K rows {0,1}; 1: K rows {2,3}
    const int l16  = lane & 15;
    const int wave = (int)threadIdx.x >> 5;
    const int tile = (int)blockIdx.x * WAVES_PER_BLOCK + wave;
    if (tile * 16 >= npix) return;       // wave-uniform guard
    const int p = tile * 16 + l16;       // this lane's pixel (column N = l16)

    // B matrix (4x16, K x N): column n = f * [-2px, -2py, 1, px^2+py^2]
    const float px = pg[2 * p + 0];
    const float py = pg[2 * p + 1];
    const float pn = px * px + py * py;
    v2f b;
    b.x = half ? FSCALE        : (-2.0f * FSCALE * px);
    b.y = half ? (FSCALE * pn) : (-2.0f * FSCALE * py);

    // C operand = EPS everywhere: D = (f*dist)^2 + EPS, guaranteed >= 0
    v8f ceps;
    #pragma unroll
    for (int r = 0; r < 8; ++r) ceps[r] = EPS;

    // each sample's float4 viewed as two float2 halves, selected by address
    const float2* sl = (const float2*)spts;

    float acc0 = 0.0f, acc1 = 0.0f, acc2 = 0.0f, acc3 = 0.0f;

    #pragma unroll 4
    for (int st = 0; st < NPTS / 16; ++st) {
        float2 av = sl[(st * 16 + l16) * 2 + half];   // ds_load_b64, no cndmask
        v2f a; a.x = av.x; a.y = av.y;

        // D[m][n] = (f * ||sample_m - pixel_n||)^2 + EPS  (16x16 tile)
        v8f d = __builtin_amdgcn_wmma_f32_16x16x4_f32(
            /*neg_a=*/false, a, /*neg_b=*/false, b,
            /*c_mod=*/(short)0, ceps, /*reuse_a=*/false, /*reuse_b=*/false);

        // exp(-SHARP*dist) == exp2(-sqrt(D)); negate folds into v_exp src mod
        acc0 += __builtin_amdgcn_exp2f(-__builtin_amdgcn_sqrtf(d[0]));
        acc1 += __builtin_amdgcn_exp2f(-__builtin_amdgcn_sqrtf(d[1]));
        acc2 += __builtin_amdgcn_exp2f(-__builtin_amdgcn_sqrtf(d[2]));
        acc3 += __builtin_amdgcn_exp2f(-__builtin_amdgcn_sqrtf(d[3]));
        acc0 += __builtin_amdgcn_exp2f(-__builtin_amdgcn_sqrtf(d[4]));
        acc1 += __builtin_amdgcn_exp2f(-__builtin_amdgcn_sqrtf(d[5]));
        acc2 += __builtin_amdgcn_exp2f(-__builtin_amdgcn_sqrtf(d[6]));
        acc3 += __builtin_amdgcn_exp2f(-__builtin_amdgcn_sqrtf(d[7]));
    }

    float acc = (acc0 + acc1) + (acc2 + acc3);
    // lane L holds sample rows 0-7, lane L+16 holds rows 8-15: combine halves
    acc += __shfl_xor(acc, 16, 32);

    if (half == 0) {
        // min_dist = -log(acc)/SHARP ; out = 1 - sigmoid((SW - min_dist)*(8/SW))
        float mind = -(__builtin_amdgcn_logf(acc) * LN2) * (1.0f / SHARP);
        float z = (SW - mind) * (8.0f / SW);
        float e = __builtin_amdgcn_exp2f(-z * LOG2E);   // exp(-z)
        out[p] = e * __builtin_amdgcn_rcpf(1.0f + e);   // 1 - sigmoid(z)
    }
}

extern "C" void kernel_launch(void* const* d_in, const int* in_sizes, int n_in,
                              void* d_out, int out_size, void* d_ws, size_t ws_size,
                              hipStream_t stream) {
    (void)in_sizes; (void)n_in; (void)d_ws; (void)ws_size;
    const float* cp = (const float*)d_in[0];   // control_points (16,4,2) f32
    const float* pg = (const float*)d_in[1];   // pixel_grid (NPIX,2) f32
    float* out = (float*)d_out;                // (1,512,512) f32

    int tiles  = (out_size + 15) / 16;
    int blocks = (tiles + WAVES_PER_BLOCK - 1) / WAVES_PER_BLOCK;
    BezierGlyph_wmma_kernel<<<blocks, BLOCK, 0, stream>>>(cp, pg, out, out_size);
}